// ConvPolicy14_34497177322295
// MI455X (gfx1250) — compile-verified
//
#include <hip/hip_runtime.h>
#include <hip/hip_bf16.h>

typedef __attribute__((ext_vector_type(2))) float v2f;
typedef __attribute__((ext_vector_type(8))) float v8f;

// ---- LDS layout (dword offsets; 16B-aligned where float4 copies are used) ----
constexpr int oX    = 0;            // 93  (input vector)
constexpr int oW1   = 96;           // 144
constexpr int oB1   = 240;          // 4
constexpr int oW2   = 244;          // 36
constexpr int oB2   = 280;          // 4
constexpr int oW3   = 284;          // 36
constexpr int oB3   = 320;          // 4
constexpr int oW4   = 324;          // 36
constexpr int oB4   = 360;          // 3 (+1 pad)
constexpr int oDW1  = 364;          // 48 (36 real + 12 zero pad: CIN 3 -> 4)
constexpr int oDB1  = 412;          // 4
constexpr int oDW2  = 416;          // 96
constexpr int oDB2  = 512;          // 4
constexpr int oDW3  = 516;          // 96
constexpr int oDB3  = 612;          // 4
constexpr int oDW4  = 616;          // 288
constexpr int oDB4  = 904;          // 6 (+2 pad)
constexpr int oEXT  = 912;          // 3 (+1 pad)
constexpr int oC4   = 916;          // 3 (+1 pad)
constexpr int oD2   = 920;          // 12 dense [pos(3)][chan(4)], VALU-only consumer
// Transposed guarded feature maps: [row = 2 + pos][channel], 2 zero guard rows
// on each side (covers pad<=2). A-fragment channel pairs are contiguous -> b64.
constexpr int oCAT4 = 932;          // 11 x 16  ch 0-3: dc3 out, 4-15: jcat
constexpr int oCAT3 = 1108;         // 11 x 8   ch 0-3: dc2 upsampled, 4-7: c2
constexpr int oCAT2 = 1196;         // 7  x 8   ch 0-3: dc1, 4-7: c3
constexpr int oC1   = 1252;         // 11 x 4
constexpr int oDS   = 1296;         // 7  x 4
constexpr int oCOMB = 1324;         // 5  x 4   (chan 3 = zero pad channel)
constexpr int oOUT  = 1344;         // 7  x 6 dense [pos][chan]
constexpr int LDS_N = 1392;

__device__ __forceinline__ float fast_tanh(float x) {
  // Branch-free: 1 - 2/(e^{2x}+1); v_exp + v_rcp, exact +-1 saturation.
  const float e = __expf(2.0f * x);
  return 1.0f - __fdividef(2.0f, e + 1.0f);
}

template<int N>
__device__ __forceinline__ void cpyN(float* dst, const float* src, int lane) {
#pragma unroll
  for (int i = 0; i < (N + 31) / 32; ++i) {
    const int idx = lane + i * 32;
    if (i * 32 + 32 <= N || idx < N) dst[idx] = src[idx];
  }
}

template<int N>
__device__ __forceinline__ void cpy4(float* dst, const float* src, int lane) {
  static_assert((N & 3) == 0, "");
  const float4* s = (const float4*)src;
  float4*       d = (float4*)dst;
#pragma unroll
  for (int i = 0; i < (N / 4 + 31) / 32; ++i) {
    const int idx = lane + i * 32;
    if (i * 32 + 32 <= N / 4 || idx < N / 4) d[idx] = s[idx];
  }
}

// conv1d (or transposed conv folded to conv) as im2col GEMM on the fp32 WMMA
// pipe. K ordering is kernel-position-major (k = kk*CIN + i): per unrolled
// tile, kk and the channel base are literals; only the lane-half term hi is
// runtime and folds into one base address per operand. Features are stored
// [pos][chan] so each A fragment pair (channels i, i+1 at one position) is one
// aligned ds_load_b64; B pairs are one ds_load_2addr_b32. Guard rows remove
// all bounds checks; junk in D rows m>=LOUT / cols n>=COUT is never read.
template<int CIN, int COUT, int LOUT, int PAD, bool DECONV, bool TANH,
         int FIN, int CST, int WOFF, int BOFF, int OOFF, int OST>
__device__ __forceinline__ void conv_layer(float* sL, int lane) {
  static_assert((CIN & 3) == 0, "CIN must be a multiple of 4");
  constexpr int NT = (CIN * 3) / 4;    // K tiles of 4
  const int mn = lane & 15;            // M for A, N for B
  const int hi = (lane >> 4) << 1;     // 0 (lanes 0-15) or 2 (lanes 16-31)
  const float* fa = sL + FIN + mn * CST + hi;
  const float* fb = sL + WOFF + (DECONV ? (mn * 3 + hi * (COUT * 3))
                                        : (mn * (CIN * 3) + hi * 3));
  v8f acc = {};
#pragma unroll
  for (int t = 0; t < NT; ++t) {
    const int kk = (4 * t) / CIN;        // literal per unrolled iteration
    const int cb = 4 * t - kk * CIN;
    v2f a, b;
#pragma unroll
    for (int u = 0; u < 2; ++u) {
      a[u] = fa[(2 + kk - PAD) * CST + cb + u];               // contiguous pair
      b[u] = DECONV ? fb[(cb + u) * (COUT * 3) + (2 - kk)]    // wf[o][i][kk]=w[i][o][2-kk]
                    : fb[(cb + u) * 3 + kk];
    }
    acc = __builtin_amdgcn_wmma_f32_16x16x4_f32(
        /*neg_a=*/false, a, /*neg_b=*/false, b,
        /*c_mod=*/(short)0, acc, /*reuse_a=*/false, /*reuse_b=*/false);
  }
  if (lane < COUT) {
    const float bb = sL[BOFF + lane];
#pragma unroll
    for (int l = 0; l < LOUT; ++l) {
      float v = acc[l] + bb;
      if (TANH) v = fast_tanh(v);
      sL[OOFF + l * OST + lane] = v;    // transposed: [pos row][chan=lane]
    }
  }
}

__global__ __launch_bounds__(32)
void ConvPolicy14_kernel(const float* __restrict__ x,
                         const float* __restrict__ w1,  const float* __restrict__ b1,
                         const float* __restrict__ w2,  const float* __restrict__ b2,
                         const float* __restrict__ w3,  const float* __restrict__ b3,
                         const float* __restrict__ w4,  const float* __restrict__ b4,
                         const float* __restrict__ dw1, const float* __restrict__ db1,
                         const float* __restrict__ dw2, const float* __restrict__ db2,
                         const float* __restrict__ dw3, const float* __restrict__ db3,
                         const float* __restrict__ dw4, const float* __restrict__ db4,
                         float* __restrict__ out) {
  __shared__ float sL[LDS_N];
  const int lane = threadIdx.x;

  // Zero guarded feature region [oCAT4, oOUT) = 412 dwords = 103 float4.
  {
    float4 z4; z4.x = z4.y = z4.z = z4.w = 0.0f;
    float4* fz = (float4*)(sL + oCAT4);
#pragma unroll
    for (int i = 0; i < 4; ++i) {
      const int idx = lane + i * 32;
      if (idx < 103) fz[idx] = z4;
    }
  }
  if (lane < 12) sL[oDW1 + 36 + lane] = 0.0f;   // deconv1 padded channel weights

  // Stage inputs + parameters into LDS (single global round-trip).
  cpyN<93>(sL + oX, x, lane);
  cpy4<144>(sL + oW1,  w1,  lane); cpyN<4>(sL + oB1,  b1,  lane);
  cpy4<36> (sL + oW2,  w2,  lane); cpyN<4>(sL + oB2,  b2,  lane);
  cpy4<36> (sL + oW3,  w3,  lane); cpyN<4>(sL + oB3,  b3,  lane);
  cpy4<36> (sL + oW4,  w4,  lane); cpyN<3>(sL + oB4,  b4,  lane);
  cpy4<36> (sL + oDW1, dw1, lane); cpyN<4>(sL + oDB1, db1, lane);
  cpy4<96> (sL + oDW2, dw2, lane); cpyN<4>(sL + oDB2, db2, lane);
  cpy4<96> (sL + oDW3, dw3, lane); cpyN<4>(sL + oDB3, db3, lane);
  cpy4<288>(sL + oDW4, dw4, lane); cpyN<6>(sL + oDB4, db4, lane);
  __syncthreads();

  // ext_obs = [psi, obsd[0], obsd[5]]  (obsd = x[47:53], q = x[3:7])
  if (lane == 0) {
    const float psi = atan2f(sL[oX + 6], sL[oX + 3]) - atan2f(-sL[oX + 4], sL[oX + 5]);
    sL[oEXT + 0] = psi; sL[oEXT + 1] = sL[oX + 47]; sL[oEXT + 2] = sL[oX + 52];
  }
  // jcat channels r=0..11 into CAT4 chans 4..15 (transposed): iterate a padded
  // 12x8 index space so r,l come from shifts (l==7 is skipped).
#pragma unroll
  for (int ii = 0; ii < 3; ++ii) {
    const int i = lane + ii * 32;
    const int r = i >> 3, l = i & 7;
    if (l < 7) {
      const int f = (r < 6) ? (r * 7 + l) : ((r - 6) * 7 + l);
      const float v = (f < 2) ? 0.0f : ((r < 6) ? sL[oX + 5 + f] : sL[oX + 51 + f]);
      sL[oCAT4 + (2 + l) * 16 + 4 + r] = v;
    }
  }
  __syncthreads();

  conv_layer<12, 4, 7, 1, false, true, oCAT4 + 4, 16, oW1, oB1, oC1 + 8, 4>(sL, lane);
  __syncthreads();
  conv_layer<4, 4, 7, 1, false, true, oC1, 4, oW2, oB2, oCAT3 + 20, 8>(sL, lane);
  __syncthreads();
  // downsample: means over positions {0..2},{2..4},{4..6} of c2 (CAT3 ch 4-7)
  if (lane < 12) {
    const int c = lane & 3, j = lane >> 2;
    const float* p = sL + oCAT3 + (2 + 2 * j) * 8 + 4 + c;
    sL[oDS + (2 + j) * 4 + c] = (p[0] + p[8] + p[16]) * (1.0f / 3.0f);
  }
  __syncthreads();
  conv_layer<4, 4, 3, 1, false, true, oDS, 4, oW3, oB3, oCAT2 + 20, 8>(sL, lane);
  __syncthreads();
  conv_layer<4, 3, 1, 0, false, true, oCAT2 + 4, 8, oW4, oB4, oC4, 4>(sL, lane);
  __syncthreads();
  if (lane < 3) sL[oCOMB + 8 + lane] = sL[oC4 + lane] + sL[oEXT + lane];
  __syncthreads();
  // deconv pad p -> conv pad 2-p with flipped/transposed (zero-padded) weights
  conv_layer<4, 4, 3, 2, true, true, oCOMB, 4, oDW1, oDB1, oCAT2 + 16, 8>(sL, lane);
  __syncthreads();
  conv_layer<8, 4, 3, 1, true, true, oCAT2, 8, oDW2, oDB2, oD2, 4>(sL, lane);
  __syncthreads();
  // upsample with index table [0,0,0,1,1,2,2] into CAT3 chans 0-3
  if (lane < 28) {
    const int c = lane & 3, l = lane >> 2;
    const int s = (l < 3) ? 0 : ((l < 5) ? 1 : 2);
    sL[oCAT3 + (2 + l) * 8 + c] = sL[oD2 + s * 4 + c];
  }
  __syncthreads();
  conv_layer<8, 4, 7, 1, true, true, oCAT3, 8, oDW3, oDB3, oCAT4 + 32, 16>(sL, lane);
  __syncthreads();
  conv_layer<16, 6, 7, 1, true, false, oCAT4, 16, oDW4, oDB4, oOUT, 6>(sL, lane);
  __syncthreads();

  // acts = dc4.reshape(42)[2:]; flat f = chan*7 + pos; OUT stored [pos][chan]
#pragma unroll
  for (int ii = 0; ii < 2; ++ii) {
    const int i = lane + ii * 32;
    if (ii == 0 || i < 40) {
      const int f = i + 2;
      const int r = f / 7, c = f - r * 7;
      out[i] = sL[oOUT + c * 6 + r];
    }
  }
}

extern "C" void kernel_launch(void* const* d_in, const int* in_sizes, int n_in,
                              void* d_out, int out_size, void* d_ws, size_t ws_size,
                              hipStream_t stream) {
  (void)in_sizes; (void)n_in; (void)out_size; (void)d_ws; (void)ws_size;
  ConvPolicy14_kernel<<<1, 32, 0, stream>>>(
      (const float*)d_in[0],
      (const float*)d_in[1],  (const float*)d_in[2],
      (const float*)d_in[3],  (const float*)d_in[4],
      (const float*)d_in[5],  (const float*)d_in[6],
      (const float*)d_in[7],  (const float*)d_in[8],
      (const float*)d_in[9],  (const float*)d_in[10],
      (const float*)d_in[11], (const float*)d_in[12],
      (const float*)d_in[13], (const float*)d_in[14],
      (const float*)d_in[15], (const float*)d_in[16],
      (float*)d_out);
}